// SelfAttention_57698590654942
// MI455X (gfx1250) — compile-verified
//
#include <hip/hip_runtime.h>

// Mathematical collapse of the reference:
//   softmax over the QUERY axis followed by a sum over the query axis means
//   sum_q attn[b,q,k] == 1 for every (b,k), so
//   out[b,h] = sum_k v[b,k,h] = (sum_s x[b,s,:]) @ Wv^T + S * bv
// Only x, Wv, bv matter. This is HBM/L2-bandwidth bound on the 33.5MB read of x.

typedef __attribute__((ext_vector_type(2))) float v2f;
typedef __attribute__((ext_vector_type(8))) float v8f;

namespace {
constexpr int kH    = 256;
constexpr int kS    = 1024;
constexpr int kB    = 32;
constexpr int kSCH  = 8;           // s-chunks per batch for the partial reduction
constexpr int kRows = kS / kSCH;   // 128 s-rows per chunk
}

// Kernel 1: partial column sums of x over the sequence axis.
// grid (32 batches, 8 chunks) x 256 threads. float4 coalesced streaming loads,
// LDS combine across the 4 s-subgroups, float4 partial writes.
__global__ __launch_bounds__(256) void sa_partial_sum(const float* __restrict__ x,
                                                      float* __restrict__ part) {
  const int b  = blockIdx.x;
  const int c  = blockIdx.y;
  const int t  = threadIdx.x;
  const int j4 = (t & 63) << 2;  // feature column group: 0..252 step 4
  const int sg = t >> 6;         // s-subgroup: 0..3

  const float* p = x + ((size_t)(b * kS + c * kRows + sg)) * kH + j4;

  float ax = 0.f, ay = 0.f, az = 0.f, aw = 0.f;
#pragma unroll 8
  for (int i = 0; i < kRows / 4; ++i) {
    // speculative prefetch a few KB ahead (global_prefetch_b8); dropped
    // silently if the tail runs past the allocation (speculative TH).
    __builtin_prefetch(p + (size_t)(i * 4 + 16) * kH, 0, 0);
    const float4 v = *reinterpret_cast<const float4*>(p + (size_t)i * 4 * kH);
    ax += v.x; ay += v.y; az += v.z; aw += v.w;
  }

  __shared__ float4 sd[256];
  sd[t] = make_float4(ax, ay, az, aw);
  __syncthreads();

  if (t < 64) {
    const float4 r0 = sd[t], r1 = sd[t + 64], r2 = sd[t + 128], r3 = sd[t + 192];
    float4 o;
    o.x = r0.x + r1.x + r2.x + r3.x;
    o.y = r0.y + r1.y + r2.y + r3.y;
    o.z = r0.z + r1.z + r2.z + r3.z;
    o.w = r0.w + r1.w + r2.w + r3.w;
    *reinterpret_cast<float4*>(part + (size_t)(b * kSCH + c) * kH + j4) = o;
  }
}

// Kernel 2: fold the kSCH partials into xs[b][j] = sum_s x[b,s,j].
__global__ __launch_bounds__(256) void sa_xs_reduce(const float* __restrict__ part,
                                                    float* __restrict__ xs) {
  const int b = blockIdx.x;
  const int j = threadIdx.x;
  float s = 0.f;
#pragma unroll
  for (int c = 0; c < kSCH; ++c) s += part[(size_t)(b * kSCH + c) * kH + j];
  xs[b * kH + j] = s;
}

// Kernel 3: out[32x256] = xs[32x256] @ Wv^T + S*bv via V_WMMA_F32_16X16X4_F32.
// One wave32 per block, one 16x16 output tile per wave (2 M-tiles x 16 N-tiles).
// A 16x4 f32 layout: lanes 0-15 -> K{k0,k0+1}; lanes 16-31 -> K{k0+2,k0+3}.
// B 4x16 f32 layout mirrors it with N striped across lanes; since we need
// B = Wv^T, each lane reads a contiguous float2 from row (n0 + lane&15) of Wv.
__global__ __launch_bounds__(32) void sa_out_gemm(const float* __restrict__ xs,
                                                  const float* __restrict__ Wv,
                                                  const float* __restrict__ bv,
                                                  float* __restrict__ out) {
  const int tile = blockIdx.x;        // 0..31
  const int mi   = tile & 1;          // M tile (batch rows 0-15 / 16-31)
  const int n0   = (tile >> 1) << 4;  // N tile base (output feature)
  const int lane = threadIdx.x;       // wave32, EXEC all ones
  const int half = lane >> 4;
  const int l15  = lane & 15;
  const int ksh  = half << 1;

  const float* arow = xs + (size_t)(mi * 16 + l15) * kH;
  const float* brow = Wv + (size_t)(n0 + l15) * kH;

  v8f acc = {};
#pragma unroll 4
  for (int k0 = 0; k0 < kH; k0 += 4) {
    const v2f a  = *reinterpret_cast<const v2f*>(arow + k0 + ksh);
    const v2f bm = *reinterpret_cast<const v2f*>(brow + k0 + ksh);
    // (neg_a, A, neg_b, B, c_mod, C, reuse_a, reuse_b)
    acc = __builtin_amdgcn_wmma_f32_16x16x4_f32(false, a, false, bm,
                                                (short)0, acc, false, false);
  }

  const int   h    = n0 + l15;
  const float bias = 1024.0f * bv[h];  // S * bv
#pragma unroll
  for (int r = 0; r < 8; ++r) {
    const int orow = mi * 16 + r + (half << 3);  // D layout: M=r (lo) / r+8 (hi)
    out[orow * kH + h] = acc[r] + bias;
  }
}

extern "C" void kernel_launch(void* const* d_in, const int* in_sizes, int n_in,
                              void* d_out, int out_size, void* d_ws, size_t ws_size,
                              hipStream_t stream) {
  (void)in_sizes; (void)n_in; (void)out_size; (void)ws_size;
  const float* x  = (const float*)d_in[0];
  // d_in[1..4] = Wq,bq,Wk,bk: mathematically dead (softmax over query axis).
  const float* Wv = (const float*)d_in[5];
  const float* bv = (const float*)d_in[6];
  float* out  = (float*)d_out;
  float* part = (float*)d_ws;                  // [32][8][256] f32 = 256 KB
  float* xs   = part + kB * kSCH * kH;         // [32][256]    f32 =  32 KB

  sa_partial_sum<<<dim3(kB, kSCH), 256, 0, stream>>>(x, part);
  sa_xs_reduce<<<kB, kH, 0, stream>>>(part, xs);
  sa_out_gemm<<<kB, 32, 0, stream>>>(xs, Wv, bv, out);
}